// EuclideanCodebook_1726576854541
// MI455X (gfx1250) — compile-verified
//
#include <hip/hip_runtime.h>

typedef __attribute__((ext_vector_type(2))) float v2f;
typedef __attribute__((ext_vector_type(8))) float v8f;

#define DECAY_F   0.8f
#define EPS_F     1e-5f

// ---------------------------------------------------------------------------
// 1) Row norms: x2[n] = sum_d x[n,d]^2 ; e2[c] = sum_d embed[c,d]^2  (D==64)
// ---------------------------------------------------------------------------
__global__ void prep_norms_kernel(const float* __restrict__ x,
                                  const float* __restrict__ embed,
                                  float* __restrict__ x2n, float* __restrict__ e2n,
                                  int N, int C) {
  int t = blockIdx.x * blockDim.x + threadIdx.x;
  if (t < N) {
    const float4* p = (const float4*)(x + ((size_t)t << 6));
    float s = 0.f;
#pragma unroll
    for (int i = 0; i < 16; ++i) {
      float4 v = p[i];
      s += v.x * v.x + v.y * v.y + v.z * v.z + v.w * v.w;
    }
    x2n[t] = s;
  } else if (t < N + C) {
    int c = t - N;
    const float4* p = (const float4*)(embed + ((size_t)c << 6));
    float s = 0.f;
#pragma unroll
    for (int i = 0; i < 16; ++i) {
      float4 v = p[i];
      s += v.x * v.x + v.y * v.y + v.z * v.z + v.w * v.w;
    }
    e2n[c] = s;
  }
}

// ---------------------------------------------------------------------------
// 2) Zero scatter buffers in workspace
// ---------------------------------------------------------------------------
__global__ void zero_f32_kernel(float* __restrict__ p, int n) {
  int t = blockIdx.x * blockDim.x + threadIdx.x;
  if (t < n) p[t] = 0.f;
}

// ---------------------------------------------------------------------------
// 3) Fused f32-WMMA distance + dist output + argmax.
//    Block: 256 threads = 8 waves = 2 row-groups x 4 column-waves.
//    Block owns 32 token rows; each iteration covers 64 codebook columns
//    (4 waves x 16), shared by both row-groups (embed loads hit WGP$).
//    B fragments double-buffered: loads for chunk it+1 fly over the 16
//    back-to-back chained WMMAs of chunk it.
// ---------------------------------------------------------------------------
__device__ __forceinline__ void load_bchunk(const float* __restrict__ embed,
                                            int col, int lhalf, v2f b[16]) {
  // B[k][n]: lane holds col n = lane%16, K = 2*(lane/16) + {0,1} per step.
  const float* er = embed + (((size_t)col) << 6) + (lhalf << 1);
  __builtin_prefetch(er + (128 << 6), 0, 3);  // 2 chunks ahead, near caches
#pragma unroll
  for (int k = 0; k < 16; ++k) b[k] = *(const v2f*)(er + (k << 2));
}

__device__ __forceinline__ void wmma_chunk(const v2f a[16], const v2f b[16],
                                           const float x2r[8],
                                           const float* __restrict__ e2n,
                                           float* __restrict__ dist_out,
                                           float best[8], int bidx[8],
                                           int mbase, int C, int col, int lhalf) {
  v8f acc = {};
#pragma unroll
  for (int k = 0; k < 16; ++k)
    acc = __builtin_amdgcn_wmma_f32_16x16x4_f32(false, a[k], false, b[k],
                                                (short)0, acc, false, false);
  const float e2v = e2n[col];
#pragma unroll
  for (int r = 0; r < 8; ++r) {
    float d2 = x2r[r] - 2.0f * acc[r] + e2v;
    d2 = d2 > 0.0f ? d2 : 0.0f;
    float dv = -__builtin_sqrtf(d2);
    dist_out[((size_t)(mbase + r + (lhalf << 3))) * C + col] = dv;
    bool better = (dv > best[r]) || (dv == best[r] && col < bidx[r]);
    best[r] = better ? dv : best[r];
    bidx[r] = better ? col : bidx[r];
  }
}

__global__ __launch_bounds__(256) void vq_dist_argmax_kernel(
    const float* __restrict__ x, const float* __restrict__ embed,
    const float* __restrict__ x2n, const float* __restrict__ e2n,
    float* __restrict__ dist_out, float* __restrict__ ind_out_f,
    int* __restrict__ ind_ws, int C) {
  const int lane   = threadIdx.x & 31;
  const int wave   = threadIdx.x >> 5;
  const int lmod   = lane & 15;   // row (A) / col (B,D) within tile
  const int lhalf  = lane >> 4;   // K-half for A/B, M-half for D
  const int rowgrp = wave >> 2;   // 0/1: which 16-row tile
  const int cwave  = wave & 3;    // which 16-col sub-chunk
  const int mbase  = (blockIdx.x << 5) + (rowgrp << 4);

  // Preload A tile (WMMA 16x16x4 A layout): a[k] = x[mbase+lmod][4k+2*lhalf..+1]
  v2f a[16];
  {
    const float* xr = x + (((size_t)(mbase + lmod)) << 6) + (lhalf << 1);
#pragma unroll
    for (int k = 0; k < 16; ++k) a[k] = *(const v2f*)(xr + (k << 2));
  }

  // x^2 for the 8 rows this lane's accumulator slots map to (M = r + 8*lhalf)
  float x2r[8];
#pragma unroll
  for (int r = 0; r < 8; ++r) x2r[r] = x2n[mbase + r + (lhalf << 3)];

  float best[8];
  int   bidx[8];
#pragma unroll
  for (int r = 0; r < 8; ++r) { best[r] = -__builtin_inff(); bidx[r] = 0x7fffffff; }

  const int colbase = (cwave << 4) + lmod;
  const int nchunk  = C >> 6;  // 64 columns per block-iteration; even count

  v2f b0[16], b1[16];
  load_bchunk(embed, colbase, lhalf, b0);
  for (int it = 0; it < nchunk; it += 2) {
    load_bchunk(embed, colbase + ((it + 1) << 6), lhalf, b1);
    wmma_chunk(a, b0, x2r, e2n, dist_out, best, bidx, mbase, C,
               colbase + (it << 6), lhalf);
    if (it + 2 < nchunk)
      load_bchunk(embed, colbase + ((it + 2) << 6), lhalf, b0);
    wmma_chunk(a, b1, x2r, e2n, dist_out, best, bidx, mbase, C,
               colbase + ((it + 1) << 6), lhalf);
  }

  // Argmax reduce across the 16 lanes that share a row (xor masks <16 stay
  // inside each 16-lane half of the wave32).
#pragma unroll
  for (int r = 0; r < 8; ++r) {
    float v = best[r];
    int   i = bidx[r];
#pragma unroll
    for (int m = 8; m >= 1; m >>= 1) {
      float ov = __shfl_xor(v, m, 32);
      int   oi = __shfl_xor(i, m, 32);
      if (ov > v || (ov == v && oi < i)) { v = ov; i = oi; }
    }
    best[r] = v;
    bidx[r] = i;
  }

  // Cross-wave reduce: 4 column-waves per row-group.
  __shared__ float s_val[8][16];
  __shared__ int   s_idx[8][16];
  if (lmod == 0) {
#pragma unroll
    for (int r = 0; r < 8; ++r) {
      s_val[wave][r + (lhalf << 3)] = best[r];
      s_idx[wave][r + (lhalf << 3)] = bidx[r];
    }
  }
  __syncthreads();
  if (threadIdx.x < 32) {
    const int g   = threadIdx.x >> 4;   // row-group
    const int row = threadIdx.x & 15;
    float v = s_val[g * 4][row];
    int   i = s_idx[g * 4][row];
#pragma unroll
    for (int w = 1; w < 4; ++w) {
      float ov = s_val[g * 4 + w][row];
      int   oi = s_idx[g * 4 + w][row];
      if (ov > v || (ov == v && oi < i)) { v = ov; i = oi; }
    }
    const int m = (blockIdx.x << 5) + (g << 4) + row;
    ind_ws[m]    = i;
    ind_out_f[m] = (float)i;
  }
}

// ---------------------------------------------------------------------------
// 4) quantize[n,d] = embed[ind[n], d]   (D==64)
// ---------------------------------------------------------------------------
__global__ void quantize_gather_kernel(const float* __restrict__ embed,
                                       const int* __restrict__ ind_ws,
                                       float* __restrict__ out_q, int total) {
  int t = blockIdx.x * blockDim.x + threadIdx.x;
  if (t >= total) return;
  int n = t >> 6, d = t & 63;
  out_q[t] = embed[(((size_t)ind_ws[n]) << 6) + d];
}

// ---------------------------------------------------------------------------
// 5) Scatter EMA stats: bins[c] += 1, esum[c,:] += x[n,:]   (D==64)
// ---------------------------------------------------------------------------
__global__ void scatter_stats_kernel(const float* __restrict__ x,
                                     const int* __restrict__ ind_ws,
                                     float* __restrict__ bins,
                                     float* __restrict__ esum, int total) {
  int t = blockIdx.x * blockDim.x + threadIdx.x;
  if (t >= total) return;
  int n = t >> 6, d = t & 63;
  int c = ind_ws[n];
  if (d == 0) atomicAdd(&bins[c], 1.0f);
  atomicAdd(&esum[(((size_t)c) << 6) + d], x[t]);
}

// ---------------------------------------------------------------------------
// 6) EMA lerp: new = old + (1-decay)*(stat - old)
// ---------------------------------------------------------------------------
__global__ void ema_update_kernel(const float* __restrict__ ea_in,
                                  const float* __restrict__ esum,
                                  const float* __restrict__ cs_in,
                                  const float* __restrict__ bins,
                                  float* __restrict__ out_ea,
                                  float* __restrict__ out_cs, int total) {
  int t = blockIdx.x * blockDim.x + threadIdx.x;
  if (t >= total) return;
  const float lerp = 1.0f - DECAY_F;
  out_ea[t] = ea_in[t] + lerp * (esum[t] - ea_in[t]);
  if ((t & 63) == 0) {
    int c = t >> 6;
    out_cs[c] = cs_in[c] + lerp * (bins[c] - cs_in[c]);
  }
}

// ---------------------------------------------------------------------------
// 7) denom = sum_c new_cluster_size[c]   (single block)
// ---------------------------------------------------------------------------
__global__ void denom_sum_kernel(const float* __restrict__ cs_new,
                                 float* __restrict__ denom, int C) {
  __shared__ float s[256];
  float v = 0.f;
  for (int i = threadIdx.x; i < C; i += 256) v += cs_new[i];
  s[threadIdx.x] = v;
  __syncthreads();
  for (int st = 128; st > 0; st >>= 1) {
    if (threadIdx.x < st) s[threadIdx.x] += s[threadIdx.x + st];
    __syncthreads();
  }
  if (threadIdx.x == 0) *denom = s[0];
}

// ---------------------------------------------------------------------------
// 8) new_embed = new_embed_avg / laplace_smoothed(cluster_size)
// ---------------------------------------------------------------------------
__global__ void final_embed_kernel(const float* __restrict__ ea_new,
                                   const float* __restrict__ cs_new,
                                   const float* __restrict__ denom,
                                   float* __restrict__ out_e, int total, int C) {
  int t = blockIdx.x * blockDim.x + threadIdx.x;
  if (t >= total) return;
  int c = t >> 6;
  float dn = *denom;
  float cs = (cs_new[c] + EPS_F) / (dn + (float)C * EPS_F) * dn;
  out_e[t] = ea_new[t] / cs;
}

// ---------------------------------------------------------------------------
extern "C" void kernel_launch(void* const* d_in, const int* in_sizes, int n_in,
                              void* d_out, int out_size, void* d_ws, size_t ws_size,
                              hipStream_t stream) {
  const float* x     = (const float*)d_in[0];  // (1, N, 64)
  const float* embed = (const float*)d_in[1];  // (1, C, 64)
  const float* cs_in = (const float*)d_in[2];  // (1, C)
  const float* ea_in = (const float*)d_in[3];  // (1, C, 64)

  const int C = in_sizes[2];
  const int D = in_sizes[1] / C;  // == 64 (kernels assume this)
  const int N = in_sizes[0] / D;
  (void)n_in; (void)out_size; (void)ws_size;

  // Output layout (flat f32, reference return order)
  float* out      = (float*)d_out;
  float* out_q    = out;                        // N*D
  float* out_ind  = out_q + (size_t)N * D;      // N   (indices as float)
  float* out_dist = out_ind + N;                // N*C
  float* out_cs   = out_dist + (size_t)N * C;   // C
  float* out_ea   = out_cs + C;                 // C*D
  float* out_e    = out_ea + (size_t)C * D;     // C*D

  // Workspace layout
  float* ws    = (float*)d_ws;
  float* x2n   = ws;                            // N
  float* e2n   = x2n + N;                       // C
  int*   idxw  = (int*)(e2n + C);               // N
  float* bins  = (float*)(idxw + N);            // C
  float* esum  = bins + C;                      // C*D
  float* denom = esum + (size_t)C * D;          // 1

  const int nd = N * D, cd = C * D;

  prep_norms_kernel<<<(N + C + 255) / 256, 256, 0, stream>>>(x, embed, x2n, e2n, N, C);

  const int ztot = C + cd + 1;
  zero_f32_kernel<<<(ztot + 255) / 256, 256, 0, stream>>>(bins, ztot);

  vq_dist_argmax_kernel<<<N / 32, 256, 0, stream>>>(x, embed, x2n, e2n,
                                                    out_dist, out_ind, idxw, C);

  quantize_gather_kernel<<<(nd + 255) / 256, 256, 0, stream>>>(embed, idxw, out_q, nd);

  scatter_stats_kernel<<<(nd + 255) / 256, 256, 0, stream>>>(x, idxw, bins, esum, nd);

  ema_update_kernel<<<(cd + 255) / 256, 256, 0, stream>>>(ea_in, esum, cs_in, bins,
                                                          out_ea, out_cs, cd);

  denom_sum_kernel<<<1, 256, 0, stream>>>(out_cs, denom, C);

  final_embed_kernel<<<(cd + 255) / 256, 256, 0, stream>>>(out_ea, out_cs, denom,
                                                           out_e, cd, C);
}